// PAc_70016556859886
// MI455X (gfx1250) — compile-verified
//
#include <hip/hip_runtime.h>
#include <cmath>
#include <stdint.h>

// Piecewise tanh-table lookup, fp32 -> fp32.
//   idx = clamp(floor(x*8 + 32), 0, 63);  mid = table[idx]
//   out = (x <= -4 || x >= 4) ? tanh(x) : mid
// Memory-bound: ~1.07 GB total traffic -> ~46 us floor at 23.3 TB/s.

typedef float v4f __attribute__((ext_vector_type(4)));

#define LUT_N   64
#define X_LOW  (-4.0f)
#define X_HIGH ( 4.0f)
#define MULT   ( 8.0f)   // N / (x_high - x_low)
#define ADDC   (32.0f)   // x_low * N / (x_low - x_high)

#if __has_builtin(__builtin_amdgcn_tanhf)
// gfx1250 hardware V_TANH_F32 (TRANS unit): cheap enough to evaluate
// unconditionally and select, keeping the hot loop branch-free.
#define TANH_IS_CHEAP 1
static __device__ __forceinline__ float fast_tanh(float v) {
    return __builtin_amdgcn_tanhf(v);
}
#else
#define TANH_IS_CHEAP 0
static __device__ __forceinline__ float fast_tanh(float v) {
    return tanhf(v);
}
#endif

#if __has_builtin(__builtin_amdgcn_global_load_async_to_lds_b32) && \
    __has_builtin(__builtin_amdgcn_s_wait_asynccnt)
#define HAVE_ASYNC_LDS 1
typedef __attribute__((address_space(1))) int* gbl_i32_ptr;
typedef __attribute__((address_space(3))) int* lds_i32_ptr;
#else
#define HAVE_ASYNC_LDS 0
#endif

static __device__ __forceinline__ float map_one(float xv, const float* __restrict__ lut) {
    float t = fmaf(xv, MULT, ADDC);
    int idx = (int)floorf(t);
    idx = idx < 0 ? 0 : (idx > (LUT_N - 1) ? (LUT_N - 1) : idx);   // v_med3-able clamp
    float mid = lut[idx];                                          // ds_load_b32, conflict-free
    bool tail = (xv <= X_LOW) | (xv >= X_HIGH);
#if TANH_IS_CHEAP
    float tv = fast_tanh(xv);
    return tail ? tv : mid;                                        // v_cndmask
#else
    if (tail) return fast_tanh(xv);                                // rare (~18%) divergent path
    return mid;
#endif
}

__global__ __launch_bounds__(256) void tanh_lut_stream_kernel(
    const float* __restrict__ x,
    const float* __restrict__ table,
    float* __restrict__ out,
    long long n)
{
    // Stage the 64-entry LUT into LDS: 64 dwords == exactly one per LDS bank,
    // so the per-lane gather below can never bank-conflict.
    __shared__ float lut[LUT_N];
    const int tid = threadIdx.x;

#if HAVE_ASYNC_LDS
    // CDNA5 async global->LDS DMA: one instruction, no VGPR round trip.
    // Tracked by ASYNCcnt; drain with s_wait_asynccnt 0 before the barrier.
    // Pointer casts go through uintptr_t: for LDS, the low 32 bits of a flat
    // address are exactly the AS(3) offset (ISA 10.2 aperture mapping).
    if (tid < LUT_N) {
        gbl_i32_ptr gsrc = (gbl_i32_ptr)(uintptr_t)(table + tid);
        lds_i32_ptr ldst = (lds_i32_ptr)(uintptr_t)(&lut[tid]);
        __builtin_amdgcn_global_load_async_to_lds_b32(gsrc, ldst,
                                                      /*offset=*/0, /*cpol=*/0);
    }
    __builtin_amdgcn_s_wait_asynccnt(0);
    __syncthreads();
#else
    if (tid < LUT_N) lut[tid] = table[tid];
    __syncthreads();
#endif

    const long long n4     = n >> 2;
    const long long stride = (long long)gridDim.x * blockDim.x;
    long long       i      = (long long)blockIdx.x * blockDim.x + tid;

    const v4f* __restrict__ x4 = (const v4f*)x;
    v4f*       __restrict__ o4 = (v4f*)out;

    for (; i < n4; i += stride) {
        // global_load_b128 with TH=NT: use-once stream, don't pollute L2.
        v4f v = __builtin_nontemporal_load(&x4[i]);
        v4f r;
        r.x = map_one(v.x, lut);
        r.y = map_one(v.y, lut);
        r.z = map_one(v.z, lut);
        r.w = map_one(v.w, lut);
        // global_store_b128 with TH=NT.
        __builtin_nontemporal_store(r, &o4[i]);
    }

    // Scalar remainder (n % 4), handled by block 0 only.
    const long long tail_base = n4 << 2;
    if (blockIdx.x == 0) {
        long long j = tail_base + tid;
        if (j < n) {
            out[j] = map_one(__builtin_nontemporal_load(&x[j]), lut);
        }
    }
}

extern "C" void kernel_launch(void* const* d_in, const int* in_sizes, int n_in,
                              void* d_out, int out_size, void* d_ws, size_t ws_size,
                              hipStream_t stream) {
    (void)n_in; (void)out_size; (void)d_ws; (void)ws_size;

    const float* x     = (const float*)d_in[0];
    const float* table = (const float*)d_in[1];
    float*       out   = (float*)d_out;

    const long long n  = (long long)in_sizes[0];
    const long long n4 = n >> 2;

    const int block = 256;                       // 8 wave32 waves per workgroup
    long long blocks = (n4 + block - 1) / block; // one float4 per thread, full cover
    if (blocks < 1) blocks = 1;
    if (blocks > 2147483647LL) blocks = 2147483647LL;

    tanh_lut_stream_kernel<<<(unsigned int)blocks, block, 0, stream>>>(x, table, out, n);
}